// EMAPositionSmoother_14783277433315
// MI455X (gfx1250) — compile-verified
//
#include <hip/hip_runtime.h>

// EMA over (64, 8192, 51) fp32: y[0]=x[0]; y[t]=0.25*x[t]+0.75*y[t-1].
// Memory-bound (214 MB min traffic -> ~9.2us @ 23.3 TB/s). Implemented as a
// 3-pass chunked parallel scan (exact linear-recurrence algebra) so HBM can
// be saturated, with the 16-step in-tile scan expressed as Y = T*X via
// V_WMMA_F32_16X16X4_F32 (T = constant lower-triangular EMA operator).

typedef float v2f __attribute__((ext_vector_type(2)));
typedef float v8f __attribute__((ext_vector_type(8)));

#define BATCH 64
#define SEQ   8192
#define FEAT  51
#define NCH   64          // chunks along time
#define CHL   (SEQ/NCH)   // 128 steps per chunk
#define TAIL  80          // 0.75^80 ~ 1e-10: tail suffices for chunk endpoint
#define FP    64          // padded feature stride in workspace
#define AEMA  0.25f
#define CEMA  0.75f

__device__ __forceinline__ float cpow(int e) {
  float r = 1.0f;
  for (int i = 0; i < e; ++i) r *= CEMA;
  return r;
}

// ---- Kernel 1: per-chunk local EMA endpoint (reads only the decayed tail) ---
__global__ void ema_chunk_tail(const float* __restrict__ x,
                               float* __restrict__ loc) {
  const int ch = blockIdx.x;          // chunk
  const int b  = blockIdx.y;          // batch
  const int f  = threadIdx.x;         // 0..63 (>=51 padded lanes, clamped)
  const int fe = (f < FEAT) ? f : (FEAT - 1);
  const float* p = x + ((size_t)b * SEQ + (size_t)ch * CHL + (CHL - TAIL)) * FEAT + fe;
  float acc = 0.0f;
#pragma unroll 4
  for (int j = 0; j < TAIL; ++j) {
    acc = __builtin_fmaf(CEMA, acc, AEMA * p[0]);   // running local EMA
    p += FEAT;
  }
  loc[((b * NCH) + ch) * FP + f] = acc;
}

// ---- Kernel 2: propagate carries across chunks (tiny, exact) ---------------
__global__ void ema_carry_scan(const float* __restrict__ loc,
                               float* __restrict__ cin) {
  const int b = blockIdx.x;
  const int f = threadIdx.x;
  const float cL = cpow(CHL);         // 0.75^128 ~ 1.04e-16 (normal fp32)
  float prev = 0.0f;                  // carry into chunk 0 is 0 (y0=x0 in T0)
  for (int k = 0; k < NCH; ++k) {
    const int idx = (b * NCH + k) * FP + f;
    cin[idx] = prev;
    prev = __builtin_fmaf(cL, prev, loc[idx]);
  }
}

// ---- Kernel 3: full scan per chunk via WMMA tiles, carry-corrected ---------
// Block = 128 threads = 4 waves; wave w handles feature tile [16w, 16w+16).
__global__ void ema_scan_wmma(const float* __restrict__ x,
                              const float* __restrict__ cin,
                              float* __restrict__ y) {
  const int ch   = blockIdx.x;
  const int b    = blockIdx.y;
  const int lane = threadIdx.x & 31;
  const int wv   = threadIdx.x >> 5;      // 0..3
  const int n    = lane & 15;             // N (feature-in-tile) / M for A
  const int hi   = lane >> 4;             // lane group
  const int f    = wv * 16 + n;
  const int fe   = (f < FEAT) ? f : (FEAT - 1);

  // A = T in ISA 16x4 f32 layout: element(M,K): lane = M + 16*(K>=2), vgpr=K%2
  float aM[4][2];
#pragma unroll
  for (int k = 0; k < 4; ++k)
#pragma unroll
    for (int v = 0; v < 2; ++v) {
      const int Kg = 4 * k + v + 2 * hi;
      aM[k][v] = (n >= Kg) ? AEMA * cpow(n - Kg) : 0.0f;
    }
  const float a00_first = cpow(n);        // T0 column 0: weight c^i (y0=x0)

  // carry application powers: element(M,N) gets +c^{M+1}*carry, M = r + 8*hi
  float cpw[8];
#pragma unroll
  for (int r = 0; r < 8; ++r) cpw[r] = cpow(r + 8 * hi + 1);

  float carry = cin[((b * NCH) + ch) * FP + wv * 16 + n];

  for (int tt = 0; tt < CHL / 16; ++tt) {
    const int Tb = ch * CHL + tt * 16;
    const bool firstTile = (ch == 0) && (tt == 0);

    // B = X tile in ISA 4x16 layout: element(K,N): lane = N + 16*(K>=2), vgpr=K%2
    const float* px = x + ((size_t)b * SEQ + Tb + 2 * hi) * FEAT + fe;
    v2f Bm[4];
#pragma unroll
    for (int k = 0; k < 4; ++k) {
      Bm[k].x = px[(4 * k) * FEAT];
      Bm[k].y = px[(4 * k + 1) * FEAT];
    }

    v8f d = {0.f, 0.f, 0.f, 0.f, 0.f, 0.f, 0.f, 0.f};
#if __has_builtin(__builtin_amdgcn_wmma_f32_16x16x4_f32)
#pragma unroll
    for (int k = 0; k < 4; ++k) {
      v2f Am;
      Am.x = (k == 0 && firstTile && hi == 0) ? a00_first : aM[k][0];
      Am.y = aM[k][1];
      d = __builtin_amdgcn_wmma_f32_16x16x4_f32(
          /*neg_a=*/false, Am, /*neg_b=*/false, Bm[k],
          /*c_mod=*/(short)0, d, /*reuse_a=*/false, /*reuse_b=*/false);
    }
#else
    // Exact VALU fallback: serial 16-step local scan, keep rows M in [8hi,8hi+8)
    {
      float run = 0.0f;
      for (int j = 0; j < 16; ++j) {
        const float xv = x[((size_t)b * SEQ + Tb + j) * FEAT + fe];
        run = (firstTile && j == 0) ? xv : __builtin_fmaf(CEMA, run, AEMA * xv);
        const int rr = j - 8 * hi;
        if (rr >= 0 && rr < 8) d[rr] = run;
      }
    }
#endif

    // Carry-correct and store: element(M,N) at vgpr r: M = r + 8*hi, N = n.
    const float last = d[7] + cpw[7] * carry;   // row M=15 lives in lanes 16..31
#pragma unroll
    for (int r = 0; r < 8; ++r) {
      const float val = d[r] + cpw[r] * carry;
      const int Mg = r + 8 * hi;
      if (f < FEAT)
        y[((size_t)b * SEQ + Tb + Mg) * FEAT + f] = val;
    }
    carry = __shfl(last, 16 + n, 32);           // broadcast y[tile_end] per N
  }
}

extern "C" void kernel_launch(void* const* d_in, const int* in_sizes, int n_in,
                              void* d_out, int out_size, void* d_ws, size_t ws_size,
                              hipStream_t stream) {
  (void)in_sizes; (void)n_in; (void)out_size; (void)ws_size;
  const float* x = (const float*)d_in[0];
  float* y = (float*)d_out;
  float* loc = (float*)d_ws;                       // 64*64*64 floats
  float* cin = loc + (size_t)BATCH * NCH * FP;     // 64*64*64 floats (2 MB total)

  ema_chunk_tail<<<dim3(NCH, BATCH), 64, 0, stream>>>(x, loc);
  ema_carry_scan<<<dim3(BATCH), 64, 0, stream>>>(loc, cin);
  ema_scan_wmma<<<dim3(NCH, BATCH), 128, 0, stream>>>(x, cin, y);
}